// ProbAttention_4853313045192
// MI455X (gfx1250) — compile-verified
//
#include <hip/hip_runtime.h>
#include <hip/hip_bf16.h>
#include <math.h>

typedef float v2f __attribute__((ext_vector_type(2)));
typedef float v4f __attribute__((ext_vector_type(4)));
typedef float v8f __attribute__((ext_vector_type(8)));

#define D_DIM 64
#define NEGBIG (-3.0e38f)

// ---------------- DPP 16-lane butterfly reductions (pure VALU, no DS) -------
template <int CTRL>
__device__ __forceinline__ float dppmov(float x) {
  return __builtin_bit_cast(float, __builtin_amdgcn_update_dpp(
      0, __builtin_bit_cast(int, x), CTRL, 0xF, 0xF, true));
}
template <int CTRL>
__device__ __forceinline__ int dppmov_i(int x) {
  return __builtin_amdgcn_update_dpp(0, x, CTRL, 0xF, 0xF, true);
}
// after these 4 stages every lane of each 16-lane row holds the reduction
__device__ __forceinline__ float redmax16(float x) {
  x = fmaxf(x, dppmov<0xB1>(x));   // quad_perm [1,0,3,2]  (xor 1)
  x = fmaxf(x, dppmov<0x4E>(x));   // quad_perm [2,3,0,1]  (xor 2)
  x = fmaxf(x, dppmov<0x141>(x));  // row_half_mirror      (quads uniform -> xor 4)
  x = fmaxf(x, dppmov<0x140>(x));  // row_mirror           (octs uniform  -> xor 8)
  return x;
}
__device__ __forceinline__ float redsum16(float x) {
  x += dppmov<0xB1>(x);
  x += dppmov<0x4E>(x);
  x += dppmov<0x141>(x);
  x += dppmov<0x140>(x);
  return x;
}
__device__ __forceinline__ int redmaxi16(int x) {
  int y;
  y = dppmov_i<0xB1>(x);  x = x > y ? x : y;
  y = dppmov_i<0x4E>(x);  x = x > y ? x : y;
  y = dppmov_i<0x141>(x); x = x > y ? x : y;
  y = dppmov_i<0x140>(x); x = x > y ? x : y;
  return x;
}

// ---------------------------------------------------------------------------
// Kernel 1: M[b,q] = max_s(q.k_s) - sum_s(q.k_s)/L
// One HALF-WAVE (16 lanes x float4 = 64 features) per (b,q); DPP reductions.
// ---------------------------------------------------------------------------
__global__ void k_mscore(const float* __restrict__ Q, const float* __restrict__ K,
                         const int* __restrict__ IDX, float* __restrict__ M,
                         int B, int L, int S) {
  int wid  = (blockIdx.x * blockDim.x + threadIdx.x) >> 5;
  int lane = threadIdx.x & 31;
  int hk = lane >> 4, l16 = lane & 15;
  int bq = wid * 2 + hk;                 // linear (b*L + q)
  if (bq >= B * L) return;
  int qi = bq % L;
  const float* kbat = K + (size_t)(bq / L) * L * D_DIM;

  v4f qv = *(const v4f*)(Q + (size_t)bq * D_DIM + l16 * 4);

  float mx = -INFINITY, sm = 0.f;
#pragma unroll 4
  for (int s = 0; s < S; ++s) {
    int ki = IDX[qi * S + s];
    v4f kv = *(const v4f*)(kbat + (size_t)ki * D_DIM + l16 * 4);
    float p = qv[0] * kv[0] + qv[1] * kv[1] + qv[2] * kv[2] + qv[3] * kv[3];
    p = redsum16(p);
    mx = fmaxf(mx, p);
    sm += p;
  }
  if (l16 == 0) M[bq] = mx - sm / (float)L;
}

// ---------------------------------------------------------------------------
// Kernel 2: per-batch top-u via iterative argmax over an LDS copy
// ---------------------------------------------------------------------------
__global__ void k_topk(const float* __restrict__ M, int* __restrict__ MTOP,
                       int L, int u) {
  __shared__ float vals[2048];
  __shared__ float bestv[256];
  __shared__ int   besti[256];
  int b = blockIdx.x, tid = threadIdx.x, nthr = blockDim.x;

  for (int i = tid; i < L; i += nthr) vals[i] = M[(size_t)b * L + i];
  __syncthreads();

  for (int t = 0; t < u; ++t) {
    float bv = -INFINITY; int bi = 0;
    for (int i = tid; i < L; i += nthr) {
      float v = vals[i];
      if (v > bv) { bv = v; bi = i; }
    }
    bestv[tid] = bv; besti[tid] = bi;
    __syncthreads();
    for (int s = nthr >> 1; s > 0; s >>= 1) {
      if (tid < s && bestv[tid + s] > bestv[tid]) {
        bestv[tid] = bestv[tid + s];
        besti[tid] = besti[tid + s];
      }
      __syncthreads();
    }
    if (tid == 0) { MTOP[b * u + t] = besti[0]; vals[besti[0]] = -INFINITY; }
    __syncthreads();
  }
}

// ---------------------------------------------------------------------------
// Kernel 3a/3b: chunked cumsum of values along L (thread = channel d)
// ---------------------------------------------------------------------------
__global__ void k_csum_part(const float* __restrict__ V, float* __restrict__ part,
                            int L, int NC) {
  int c = blockIdx.x % NC, b = blockIdx.x / NC, d = threadIdx.x;
  int CL = L / NC;
  const float* vb = V + ((size_t)b * L + (size_t)c * CL) * D_DIM + d;
  float s = 0.f;
  for (int l = 0; l < CL; ++l) s += vb[(size_t)l * D_DIM];
  part[((size_t)b * NC + c) * D_DIM + d] = s;
}

__global__ void k_csum_scan(const float* __restrict__ V, const float* __restrict__ part,
                            float* __restrict__ OUT, int L, int NC) {
  int c = blockIdx.x % NC, b = blockIdx.x / NC, d = threadIdx.x;
  int CL = L / NC;
  float acc = 0.f;
  for (int cc = 0; cc < c; ++cc) acc += part[((size_t)b * NC + cc) * D_DIM + d];
  const float* vb = V + ((size_t)b * L + (size_t)c * CL) * D_DIM + d;
  float* ob = OUT + ((size_t)b * L + (size_t)c * CL) * D_DIM + d;
  for (int l = 0; l < CL; ++l) { acc += vb[(size_t)l * D_DIM]; ob[(size_t)l * D_DIM] = acc; }
}

// ---------------------------------------------------------------------------
// Kernel 4: split-K WMMA flash attention partials.
// Grid = B*NS; block = UT waves; wave = 16-query tile over CHUNK=L/NS keys.
// __launch_bounds__(128,1): full VGPR budget so nothing spills to scratch.
//   A layout (16x4):  lane = 16*(K/2)+M, vgpr j -> K = 2*(lane/16)+j
//   B layout (4x16):  lane = 16*(K/2)+N, vgpr j -> K = j + 2*(lane/16)
//   C layout (16x16): vgpr r, lane -> (M = r + 8*(lane/16), N = lane%16)
// ---------------------------------------------------------------------------
__global__ void __launch_bounds__(128, 1)
k_attn_split(const float* __restrict__ Q, const float* __restrict__ K,
             const float* __restrict__ V, const int* __restrict__ MTOP,
             float* __restrict__ PO, float* __restrict__ PML,
             int L, int u, int NS, int ROWS) {
  __shared__ float pbuf[4][16][17];   // per-wave 16x16 P tile (+pad)

  int b    = blockIdx.x / NS;
  int sp   = blockIdx.x % NS;
  int wave = threadIdx.x >> 5;
  int lane = threadIdx.x & 31;
  int n    = lane & 15;
  int hk   = lane >> 4;
  const int CHUNK = L / NS;
  const int kb0   = sp * CHUNK;

  // A-layout query row for this lane (clamped; padded rows never stored)
  int posA = wave * 16 + n; if (posA > u - 1) posA = u - 1;
  int qa = MTOP[b * u + posA];
  int tilemax = redmaxi16(qa);         // max selected position in this 16-row tile

  // C-layout row metadata (8 rows per lane)
  int gq[8], valid[8];
  for (int r = 0; r < 8; ++r) {
    int pr = wave * 16 + r + 8 * hk;
    valid[r] = (pr < u);
    gq[r] = MTOP[b * u + (valid[r] ? pr : (u - 1))];
  }

  // Preload Q tile in A layout
  v2f aq[16];
  const float* qbase = Q + ((size_t)b * L + qa) * D_DIM;
#pragma unroll
  for (int t = 0; t < 16; ++t) aq[t] = *(const v2f*)(qbase + t * 4 + hk * 2);

  float mrow[8], lrow[8];
  v8f acc[4] = {};
  for (int r = 0; r < 8; ++r) { mrow[r] = NEGBIG; lrow[r] = 0.f; }

  const float scale = 0.125f;          // 1/sqrt(64)
  const int NT = CHUNK >> 4;

  for (int kt = 0; kt < NT; ++kt) {
    const int kb = kb0 + kt * 16;
    if (kb > tilemax) break;           // wave-uniform: all later tiles fully masked

    // ---- batched K-tile load (one clause, one wait) -----------------------
    const float* kbase = K + ((size_t)b * L + kb + n) * D_DIM;
    v2f bk[16];
#pragma unroll
    for (int t = 0; t < 16; ++t) bk[t] = *(const v2f*)(kbase + t * 4 + hk * 2);

    // ---- S = Q.K^T : two independent 8-deep WMMA chains for ILP -----------
    v8f sa = {}, sb = {};
#pragma unroll
    for (int t = 0; t < 8; ++t) {
      sa = __builtin_amdgcn_wmma_f32_16x16x4_f32(false, aq[t], false, bk[t],
                                                 (short)0, sa, false, false);
      sb = __builtin_amdgcn_wmma_f32_16x16x4_f32(false, aq[t + 8], false, bk[t + 8],
                                                 (short)0, sb, false, false);
    }
    v8f s = sa + sb;

    // ---- preload V tile now so its latency hides under softmax VALU -------
    v2f bv[4][4];
    const float* vbase = V + ((size_t)b * L + kb) * D_DIM + n;
#pragma unroll
    for (int tp = 0; tp < 4; ++tp)
#pragma unroll
      for (int j = 0; j < 4; ++j) {
        const float* p0 = vbase + (size_t)(tp * 4 + hk * 2) * D_DIM + j * 16;
        v2f t2; t2[0] = p0[0]; t2[1] = p0[D_DIM];
        bv[tp][j] = t2;
      }

    // ---- scale + mask + online softmax (DPP reductions, NaN-free) ---------
    const int keyc = kb + n;
    float p[8], corr[8];
#pragma unroll
    for (int r = 0; r < 8; ++r) {
      bool masked = keyc > gq[r];
      float sv = masked ? NEGBIG : s[r] * scale;
      float m2 = fmaxf(mrow[r], redmax16(sv));
      corr[r] = __expf(mrow[r] - m2);            // args finite: no NaN
      float pv = masked ? 0.f : __expf(sv - m2); // sv <= m2
      lrow[r] = lrow[r] * corr[r] + redsum16(pv);
      mrow[r] = m2;
      p[r] = pv;
    }
#pragma unroll
    for (int j = 0; j < 4; ++j)
#pragma unroll
      for (int r = 0; r < 8; ++r) acc[j][r] *= corr[r];

    // ---- transpose P (C layout -> A layout) through per-wave LDS tile -----
#pragma unroll
    for (int r = 0; r < 8; ++r) pbuf[wave][r + 8 * hk][n] = p[r];
    asm volatile("s_wait_dscnt 0" ::: "memory");  // wave-local LDS RAW

    // ---- O += P.V : 16 WMMAs (4 independent acc chains) -------------------
#pragma unroll
    for (int tp = 0; tp < 4; ++tp) {
      v2f pa;
      pa[0] = pbuf[wave][n][tp * 4 + hk * 2];
      pa[1] = pbuf[wave][n][tp * 4 + hk * 2 + 1];
#pragma unroll
      for (int j = 0; j < 4; ++j)
        acc[j] = __builtin_amdgcn_wmma_f32_16x16x4_f32(false, pa, false, bv[tp][j],
                                                       (short)0, acc[j], false, false);
    }
  }

  // ---- write unnormalized partials --------------------------------------
  for (int r = 0; r < 8; ++r) {
    if (!valid[r]) continue;
    int row = wave * 16 + r + 8 * hk;
    size_t base = ((size_t)b * NS + sp) * ROWS + row;
    float* po = PO + base * D_DIM + n;
#pragma unroll
    for (int j = 0; j < 4; ++j) po[j * 16] = acc[j][r];
    if (n == 0) { PML[base * 2] = mrow[r]; PML[base * 2 + 1] = lrow[r]; }
  }
}

// ---------------------------------------------------------------------------
// Kernel 5: combine split-K partials; write normalized rows into OUT
// ---------------------------------------------------------------------------
__global__ void k_combine(const float* __restrict__ PO, const float* __restrict__ PML,
                          const int* __restrict__ MTOP, float* __restrict__ OUT,
                          int L, int u, int NS, int ROWS) {
  int b = blockIdx.x / u, row = blockIdx.x % u, d = threadIdx.x;
  float M = NEGBIG;
  for (int s = 0; s < NS; ++s)
    M = fmaxf(M, PML[(((size_t)b * NS + s) * ROWS + row) * 2]);
  float Ltot = 0.f, O = 0.f;
  for (int s = 0; s < NS; ++s) {
    size_t base = ((size_t)b * NS + s) * ROWS + row;
    float m = PML[base * 2], l = PML[base * 2 + 1];
    float w = __expf(m - M);
    Ltot += w * l;
    O += w * PO[base * D_DIM + d];
  }
  int gq = MTOP[b * u + row];
  OUT[((size_t)b * L + gq) * D_DIM + d] = O / Ltot;
}

// ---------------------------------------------------------------------------
extern "C" void kernel_launch(void* const* d_in, const int* in_sizes, int n_in,
                              void* d_out, int out_size, void* d_ws, size_t ws_size,
                              hipStream_t stream) {
  const float* Q  = (const float*)d_in[0];
  const float* K  = (const float*)d_in[1];
  const float* V  = (const float*)d_in[2];
  const int* IDX  = (const int*)d_in[3];
  float* OUT = (float*)d_out;

  const int L = 2048, D = D_DIM;
  const int B = in_sizes[0] / (L * D);
  const int S = in_sizes[3] / L;           // sample_k
  const int u = (S < L) ? S : L;           // 40
  const int NC = 16;                       // cumsum chunks
  const int NS = 8;                        // key splits for flash attention
  const int UT = (u + 15) / 16;            // query tiles (waves per block) <= 4
  const int ROWS = UT * 16;                // padded rows per (b,split)

  // workspace layout (floats)
  float* Mbuf = (float*)d_ws;                          // B*L
  int*   Mtop = (int*)(Mbuf + (size_t)B * L);          // B*u
  float* part = (float*)(Mtop + (size_t)B * u);        // B*NC*D
  float* PML  = part + (size_t)B * NC * D;             // B*NS*ROWS*2
  float* PO   = PML + (size_t)B * NS * ROWS * 2;       // B*NS*ROWS*D

  { // K1: M scores (one half-wave per (b,q))
    int hw = B * L;                        // half-waves needed
    int waves = (hw + 1) / 2;
    int threads = 256;
    int blocks = (waves * 32 + threads - 1) / threads;
    k_mscore<<<blocks, threads, 0, stream>>>(Q, K, IDX, Mbuf, B, L, S);
  }
  // K2: top-u per batch
  k_topk<<<B, 256, 0, stream>>>(Mbuf, Mtop, L, u);
  // K3: cumsum baseline into OUT
  k_csum_part<<<B * NC, D, 0, stream>>>(V, part, L, NC);
  k_csum_scan<<<B * NC, D, 0, stream>>>(V, part, OUT, L, NC);
  // K4: split-K WMMA attention partials
  k_attn_split<<<B * NS, 32 * UT, 0, stream>>>(Q, K, V, Mtop, PO, PML, L, u, NS, ROWS);
  // K5: combine + scatter into OUT
  k_combine<<<B * u, D, 0, stream>>>(PO, PML, Mtop, OUT, L, u, NS, ROWS);
}